// PseudoLabelPlus_61581241090537
// MI455X (gfx1250) — compile-verified
//
#include <hip/hip_runtime.h>
#include <math.h>

#define SS 128
#define BB 2048
#define VV 50000
#define EE 300
#define FF 9
#define KK 10
#define FP1 10  // F+1

typedef __attribute__((ext_vector_type(2))) float v2f;
typedef __attribute__((ext_vector_type(8))) float v8f;
typedef unsigned int u32x4 __attribute__((ext_vector_type(4)));
typedef int i32x4 __attribute__((ext_vector_type(4)));
typedef int i32x8 __attribute__((ext_vector_type(8)));

#if __has_builtin(__builtin_amdgcn_tensor_load_to_lds) && __has_builtin(__builtin_amdgcn_s_wait_tensorcnt)
#define HAVE_TDM 1
#else
#define HAVE_TDM 0
#endif

#if HAVE_TDM
// Issue one TDM gather: 16 embedding rows (token ids at text[(s0+i)*B + b]) -> LDS.
// Uniform (SALU/SMEM) descriptor build; caller must be wave 0 only.
__device__ __forceinline__ void tdm_gather16(const int* __restrict__ text, int b, int s0,
                                             unsigned lds_byte,
                                             const float* __restrict__ emb_table)
{
    const unsigned long long ga = (unsigned long long)(uintptr_t)emb_table;
    u32x4 g0;
    g0[0] = 0x80000001u;                                         // count=1 | gather_mode=1 (idx16)
    g0[1] = lds_byte;                                            // lds_addr (bytes)
    g0[2] = (unsigned)ga;                                        // global_addr[31:0]
    g0[3] = (unsigned)((ga >> 32) & 0x01FFFFFFu) | (2u << 30);   // global_addr[56:32] | type=2
    i32x8 g1;
    g1[0] = (2 << 16);                  // data_size = 4 bytes; no multicast/pad/iterate
    g1[1] = (EE << 16);                 // tensor_dim0 = 300 (bits 79:48, low16 here)
    g1[2] = (int)((unsigned)VV << 16);  // tensor_dim0 hi=0 | tensor_dim1 = 50000 (low16)
    g1[3] = (EE << 16);                 // tensor_dim1 hi=0 | tile_dim0 = 300
    g1[4] = 16;                         // tile_dim1 = #valid gather indices
    g1[5] = EE;                         // tensor_dim0_stride = 300 (low 32)
    g1[6] = 0;                          // stride hi | tensor_dim1_stride (ignored in gather)
    g1[7] = 0;
    int t[16];
    #pragma unroll
    for (int i = 0; i < 16; ++i) t[i] = text[(s0 + i) * BB + b];
    i32x4 g2, g3;
    #pragma unroll
    for (int i = 0; i < 4; ++i) g2[i] = (t[2 * i] & 0xFFFF) | (t[2 * i + 1] << 16);
    #pragma unroll
    for (int i = 0; i < 4; ++i) g3[i] = (t[8 + 2 * i] & 0xFFFF) | (t[8 + 2 * i + 1] << 16);
    // clang-23 form takes a trailing int32x8 before cpol; semantics uncharacterized.
    // Mirror groups 2+3 into it so either hardware-group mapping sees the indices.
    i32x8 gx;
    #pragma unroll
    for (int i = 0; i < 4; ++i) { gx[i] = g2[i]; gx[i + 4] = g3[i]; }
    __builtin_amdgcn_tensor_load_to_lds(g0, g1, g2, g3, gx, 0);
}
#endif

// One 64-thread block (2 waves) per batch element b.
__global__ __launch_bounds__(64) void plp_main(
    const int*   __restrict__ text,       // [S,B]
    const float* __restrict__ emb_table,  // [V,E]
    const float* __restrict__ conv_w,     // [F,E]
    const float* __restrict__ conv_b,     // [F]
    const float* __restrict__ Wk,         // [E,K]
    const float* __restrict__ bk,         // [K]
    float*       __restrict__ qk_out)     // [B,F+1]
{
#if HAVE_TDM
    __shared__ float emb_lds[2 * 32 * EE];  // double-buffered 32-row chunks (TDM destination)
#else
    __shared__ float emb_lds[32 * EE];
#endif
    __shared__ float cw_lds[16 * EE];      // conv_w padded to 16 rows (rows 9..15 zero)
    __shared__ float scores_lds[32 * 16];  // WMMA output per chunk
    __shared__ float sen_lds[EE + 4];
    __shared__ float mean_lds[EE + 4];
    __shared__ float w_lds[32];
    __shared__ float cb_lds[16];
    __shared__ float qraw_lds[16];
    __shared__ float z_lds[16];
    __shared__ float qk_lds[16];

    const int b    = blockIdx.x;
    const int tid  = threadIdx.x;
    const int lane = tid & 31;
    const int wave = tid >> 5;
    const int half = lane >> 4;  // 0: K={0,1}, 1: K={2,3}
    const int ln   = lane & 15;

#if HAVE_TDM
    const unsigned lds0 = (unsigned)(uintptr_t)&emb_lds[0];        // chunk buffer 0
    const unsigned lds1 = (unsigned)(uintptr_t)&emb_lds[32 * EE];  // chunk buffer 1
    const unsigned hrow = 16u * EE * 4u;                           // byte offset of rows 16..31
    // Kick off chunk 0 gather (2 descriptors) from wave 0 before anything else.
    if (tid < 32) {
        tdm_gather16(text, b, 0, lds0, emb_table);
        tdm_gather16(text, b, 16, lds0 + hrow, emb_table);
    }
#endif

    // conv_w -> LDS, padded to 16 rows of zeros (same thread writes same index twice: ordered)
    for (int i = tid; i < 16 * EE; i += 64) cw_lds[i] = 0.0f;
    for (int i = tid; i < FF * EE; i += 64) cw_lds[i] = conv_w[i];
    if (tid < FF) cb_lds[tid] = conv_b[tid];

    float sen_acc[5]  = {0.f, 0.f, 0.f, 0.f, 0.f};
    float mean_acc[5] = {0.f, 0.f, 0.f, 0.f, 0.f};

    __syncthreads();

    for (int c = 0; c < 4; ++c) {
#if HAVE_TDM
        // Prefetch chunk c+1 into the other buffer, then wait for chunk c (in-order TENSORcnt).
        if (tid < 32) {
            if (c < 3) {
                const unsigned nbuf = ((c + 1) & 1) ? lds1 : lds0;
                tdm_gather16(text, b, 32 * (c + 1), nbuf, emb_table);
                tdm_gather16(text, b, 32 * (c + 1) + 16, nbuf + hrow, emb_table);
                __builtin_amdgcn_s_wait_tensorcnt(2);
            } else {
                __builtin_amdgcn_s_wait_tensorcnt(0);
            }
        }
        __syncthreads();
        const float* ebuf = emb_lds + (c & 1) * (32 * EE);
#else
        const int sbase = c * 32;
        for (int ss = 0; ss < 32; ++ss) {
            const long row = (long)text[(long)(sbase + ss) * BB + b];
            const float* src = emb_table + row * EE;
            float* dst = emb_lds + ss * EE;
            for (int e = tid; e < EE; e += 64) dst[e] = src[e];
        }
        __syncthreads();
        const float* ebuf = emb_lds;
#endif

        // ---- scores via f32 WMMA: wave w does 16 tokens at local offset 16*w ----
        {
            const int ls0 = 16 * wave;
            v8f acc = {0.f, 0.f, 0.f, 0.f, 0.f, 0.f, 0.f, 0.f};
            const float* arow = ebuf + (ls0 + ln) * EE;  // A: M=ln row
            const float* brow = cw_lds + ln * EE;        // B: N=ln column (conv_w row)
            const int koff = 2 * half;
            for (int k0 = 0; k0 < EE; k0 += 4) {
                v2f a, bm;
                a[0]  = arow[k0 + koff];
                a[1]  = arow[k0 + koff + 1];
                bm[0] = brow[k0 + koff];
                bm[1] = brow[k0 + koff + 1];
                acc = __builtin_amdgcn_wmma_f32_16x16x4_f32(
                    false, a, false, bm, (short)0, acc, false, false);
            }
            #pragma unroll
            for (int r = 0; r < 8; ++r)
                scores_lds[(ls0 + r + 8 * half) * 16 + ln] = acc[r];
        }
        __syncthreads();

        // ---- per-token attention weight: max_f relu(score + bias) ----
        if (tid < 32) {
            const float* srow = scores_lds + tid * 16;
            float wmax = 0.0f;  // relu outputs are >= 0
            #pragma unroll
            for (int f = 0; f < FF; ++f) {
                float v = srow[f] + cb_lds[f];
                v = v > 0.f ? v : 0.f;
                wmax = v > wmax ? v : wmax;
            }
            w_lds[tid] = wmax;
        }
        __syncthreads();

        // ---- e-major accumulation of sen (weighted) and mean ----
        for (int s = 0; s < 32; ++s) {
            const float w = w_lds[s];
            const float* row = ebuf + s * EE;
            float v0 = row[tid];
            float v1 = row[tid + 64];
            float v2 = row[tid + 128];
            float v3 = row[tid + 192];
            sen_acc[0] += w * v0;  mean_acc[0] += v0;
            sen_acc[1] += w * v1;  mean_acc[1] += v1;
            sen_acc[2] += w * v2;  mean_acc[2] += v2;
            sen_acc[3] += w * v3;  mean_acc[3] += v3;
        }
        if (tid < EE - 256) {  // tid < 44
            for (int s = 0; s < 32; ++s) {
                const float v4 = ebuf[s * EE + tid + 256];
                sen_acc[4]  += w_lds[s] * v4;
                mean_acc[4] += v4;
            }
        }
        __syncthreads();  // before this buffer is refilled
    }

    const float inv_s = 1.0f / (float)SS;
    sen_lds[tid]        = sen_acc[0] * inv_s;  mean_lds[tid]        = mean_acc[0] * inv_s;
    sen_lds[tid + 64]   = sen_acc[1] * inv_s;  mean_lds[tid + 64]   = mean_acc[1] * inv_s;
    sen_lds[tid + 128]  = sen_acc[2] * inv_s;  mean_lds[tid + 128]  = mean_acc[2] * inv_s;
    sen_lds[tid + 192]  = sen_acc[3] * inv_s;  mean_lds[tid + 192]  = mean_acc[3] * inv_s;
    if (tid < EE - 256) {
        sen_lds[tid + 256]  = sen_acc[4] * inv_s;
        mean_lds[tid + 256] = mean_acc[4] * inv_s;
    }
    __syncthreads();

    // ---- tail dots: q_raw[f] = relu(sen . conv_w[f] + cb[f]); z[k] = mean . Wk[:,k] + bk[k]
    if (tid < FF) {
        const float* cwf = cw_lds + tid * EE;
        float acc = cb_lds[tid];
        for (int e = 0; e < EE; ++e) acc += sen_lds[e] * cwf[e];
        qraw_lds[tid] = acc > 0.f ? acc : 0.f;
    }
    if (tid >= 32 && tid < 32 + KK) {
        const int k = tid - 32;
        float acc = bk[k];
        for (int e = 0; e < EE; ++e) acc += mean_lds[e] * Wk[e * KK + k];
        z_lds[k] = acc;
    }
    __syncthreads();

    if (tid == 0) {
        float qm = qraw_lds[0];
        for (int f = 1; f < FF; ++f) qm = fmaxf(qm, qraw_lds[f]);
        float qe[FF]; float qs = 0.f;
        for (int f = 0; f < FF; ++f) { qe[f] = expf(qraw_lds[f] - qm); qs += qe[f]; }

        float zm = z_lds[0];
        for (int k = 1; k < KK; ++k) zm = fmaxf(zm, z_lds[k]);
        float ze[KK]; float zs = 0.f;
        for (int k = 0; k < KK; ++k) { ze[k] = expf(z_lds[k] - zm); zs += ze[k]; }
        float h = 0.f;
        for (int k = 0; k < KK; ++k) { float p = ze[k] / zs; h -= p * logf(p); }
        const float hn = h / logf((float)KK);
        const float qnull = (hn >= 0.5f) ? (hn - 0.5f) * 2.0f : 0.0f;  // /(1.0-0.5)
        const float scale = (1.0f - qnull) / qs;                       // softmax * (1-q_null)
        for (int j = 0; j < FP1; ++j) {
            float v;
            if (j < 4)        v = qe[j] * scale;
            else if (j == 4)  v = qnull;          // MISC_IDX
            else              v = qe[j - 1] * scale;
            qk_lds[j] = v;
        }
    }
    __syncthreads();
    if (tid < FP1) qk_out[(long)b * FP1 + tid] = qk_lds[tid];
}

// Deterministic column sums: wave j reduces column j of q_kplus over B.
__global__ __launch_bounds__(320) void plp_fs(const float* __restrict__ qk,
                                              float* __restrict__ fs)
{
    const int j    = threadIdx.x >> 5;   // 0..9
    const int lane = threadIdx.x & 31;
    float acc = 0.f;
    for (int b = lane; b < BB; b += 32) acc += qk[(long)b * FP1 + j];
    #pragma unroll
    for (int off = 16; off > 0; off >>= 1) acc += __shfl_xor(acc, off, 32);
    if (lane == 0) fs[j] = acc;
}

// p[b,:] = (qk^2 / fs) row-normalized
__global__ __launch_bounds__(128) void plp_fin(const float* __restrict__ qk,
                                               const float* __restrict__ fs,
                                               float* __restrict__ p)
{
    const int b = blockIdx.x * 128 + threadIdx.x;
    if (b >= BB) return;
    const float* row = qk + (long)b * FP1;
    float t[FP1]; float s = 0.f;
    #pragma unroll
    for (int j = 0; j < FP1; ++j) {
        const float q = row[j];
        const float v = q * q / fs[j];
        t[j] = v; s += v;
    }
    const float inv = 1.0f / s;
    float* prow = p + (long)b * FP1;
    #pragma unroll
    for (int j = 0; j < FP1; ++j) prow[j] = t[j] * inv;
}

extern "C" void kernel_launch(void* const* d_in, const int* in_sizes, int n_in,
                              void* d_out, int out_size, void* d_ws, size_t ws_size,
                              hipStream_t stream)
{
    (void)in_sizes; (void)n_in; (void)out_size; (void)ws_size;
    const int*   text = (const int*)  d_in[0];
    const float* emb  = (const float*)d_in[1];
    const float* cw   = (const float*)d_in[2];
    const float* cb   = (const float*)d_in[3];
    const float* Wk   = (const float*)d_in[4];
    const float* bk   = (const float*)d_in[5];

    float* out    = (float*)d_out;
    float* p_out  = out;             // [B, F+1]
    float* qk_out = out + BB * FP1;  // [B, F+1]
    float* fs     = (float*)d_ws;    // [F+1]

    plp_main<<<BB, 64, 0, stream>>>(text, emb, cw, cb, Wk, bk, qk_out);
    plp_fs<<<1, 320, 0, stream>>>(qk_out, fs);
    plp_fin<<<BB / 128, 128, 0, stream>>>(qk_out, fs, p_out);
}